// ScoreModel_40793599378153
// MI455X (gfx1250) — compile-verified
//
#include <hip/hip_runtime.h>
#include <math.h>

#define NN      50000
#define NGRAPH  1000
#define NEL     400000
#define NEGL    400000
#define SDIM    64
#define VDIM    16
#define NRBF    16
#define NTYPES  16
#define FDIM    96      // SDIM + 2*VDIM
#define NLAYERS 5
#define CUTOFF  5.0f

typedef float v2f __attribute__((ext_vector_type(2)));
typedef float v8f __attribute__((ext_vector_type(8)));

// ---------------- WMMA f32 16x16x4 tile helper -----------------------------
// A: 16xK row-major (lda), B: Kx16 row-major (ldb). Accumulates into acc.
// Fragment layout per CDNA5 ISA 7.12.2:
//   A 16x4 f32: lanes 0-15 hold M=lane, K={k0,k0+1}; lanes 16-31 K={k0+2,k0+3}
//   B 4x16 f32: VGPR j, lanes 0-15 = row (k0+j), lanes 16-31 = row (k0+2+j)
//   C/D: VGPR r, lanes 0-15 -> M=r, lanes 16-31 -> M=r+8, N=lane%16
// A pairs are contiguous in memory -> single b64 load per step.
__device__ __forceinline__ v8f wmma_tile_f32(const float* __restrict__ A, int lda,
                                             const float* __restrict__ B, int ldb,
                                             int K, v8f acc) {
  const int lane = threadIdx.x & 31;
  const int half = lane >> 4;
  const int l16  = lane & 15;
  for (int k0 = 0; k0 < K; k0 += 4) {
    const int ka = k0 + 2 * half;
    v2f a = *(const v2f*)(A + l16 * lda + ka);     // global_load_b64
    v2f b;
    b.x = B[(ka + 0) * ldb + l16];
    b.y = B[(ka + 1) * ldb + l16];
    acc = __builtin_amdgcn_wmma_f32_16x16x4_f32(false, a, false, b,
                                                (short)0, acc, false, false);
  }
  return acc;
}

// ---------------- edge attributes + degree count ---------------------------
__global__ void k_edge_attrs(const int* __restrict__ ei, const float* __restrict__ pos,
                             float* __restrict__ dArr, float* __restrict__ rnArr,
                             float* __restrict__ cnt, int E) {
  int e = blockIdx.x * blockDim.x + threadIdx.x;
  if (e >= E) return;
  int s = ei[e];
  int t = ei[E + e];
  float rx = pos[t * 3 + 0] - pos[s * 3 + 0];
  float ry = pos[t * 3 + 1] - pos[s * 3 + 1];
  float rz = pos[t * 3 + 2] - pos[s * 3 + 2];
  float d2 = fmaxf(rx * rx + ry * ry + rz * rz, 1e-6f);
  float d  = sqrtf(d2);
  float inv = 1.0f / d;
  dArr[e] = d;
  rnArr[e * 3 + 0] = rx * inv;
  rnArr[e * 3 + 1] = ry * inv;
  rnArr[e * 3 + 2] = rz * inv;
  atomicAdd(&cnt[t], 1.0f);
}

// ---------------- embedding stage 1: tmp = x@W_atom + b_atom + te[batch] ---
__global__ __launch_bounds__(128) void k_embed1(
    const float* __restrict__ x, const float* __restrict__ Watom,
    const float* __restrict__ batom, const float* __restrict__ t,
    const int* __restrict__ batch, const float* __restrict__ Wtime,
    const float* __restrict__ btime, float* __restrict__ tmp) {
  const int row0 = blockIdx.x * 16;
  const int ct   = threadIdx.x >> 5;       // col tile 0..3
  const int lane = threadIdx.x & 31;
  const int half = lane >> 4;
  const int l16  = lane & 15;
  const int col  = ct * 16 + l16;
  v8f acc;
  float bb = batom[col];
#pragma unroll
  for (int r = 0; r < 8; r++) acc[r] = bb;
  acc = wmma_tile_f32(x + (size_t)row0 * NTYPES, NTYPES,
                      Watom + ct * 16, SDIM, NTYPES, acc);
  float wt = Wtime[col], bt = btime[col];
#pragma unroll
  for (int r = 0; r < 8; r++) {
    int row = row0 + r + 8 * half;
    int g = batch[row];
    tmp[(size_t)row * SDIM + col] = acc[r] + t[g] * wt + bt;
  }
}

// ---------------- embedding stage 2: s = tmp@W_at + b_at -------------------
__global__ __launch_bounds__(128) void k_embed2(
    const float* __restrict__ tmp, const float* __restrict__ Wat,
    const float* __restrict__ bat, float* __restrict__ s) {
  const int row0 = blockIdx.x * 16;
  const int ct   = threadIdx.x >> 5;
  const int lane = threadIdx.x & 31;
  const int half = lane >> 4;
  const int l16  = lane & 15;
  const int col  = ct * 16 + l16;
  v8f acc;
  float bb = bat[col];
#pragma unroll
  for (int r = 0; r < 8; r++) acc[r] = bb;
  acc = wmma_tile_f32(tmp + (size_t)row0 * SDIM, SDIM, Wat + ct * 16, SDIM, SDIM, acc);
#pragma unroll
  for (int r = 0; r < 8; r++)
    s[(size_t)(row0 + r + 8 * half) * SDIM + col] = acc[r];
}

// ---------------- phi = silu(s@Wp + bp), 6 col tiles -----------------------
__global__ __launch_bounds__(192) void k_phi(
    const float* __restrict__ s, const float* __restrict__ Wp,
    const float* __restrict__ bp, float* __restrict__ phi) {
  const int row0 = blockIdx.x * 16;
  const int ct   = threadIdx.x >> 5;       // 0..5
  const int lane = threadIdx.x & 31;
  const int half = lane >> 4;
  const int l16  = lane & 15;
  const int col  = ct * 16 + l16;
  v8f acc;
  float bb = bp[col];
#pragma unroll
  for (int r = 0; r < 8; r++) acc[r] = bb;
  acc = wmma_tile_f32(s + (size_t)row0 * SDIM, SDIM, Wp + ct * 16, FDIM, SDIM, acc);
#pragma unroll
  for (int r = 0; r < 8; r++) {
    float xv = acc[r];
    float sv = xv / (1.0f + __expf(-xv));   // silu
    phi[(size_t)(row0 + r + 8 * half) * FDIM + col] = sv;
  }
}

// ---------------- per-edge message + scatter -------------------------------
// phi rows (96 floats, 384B stride) and v rows (48 floats, 192B stride) are
// 16B aligned -> gathered with float4 (global_load_b128).
__global__ __launch_bounds__(256) void k_message(
    const int* __restrict__ ei, const float* __restrict__ dArr,
    const float* __restrict__ rnArr, const float* __restrict__ phi,
    const float* __restrict__ vbuf, const float* __restrict__ Wr,
    const float* __restrict__ br, float* __restrict__ ds_agg,
    float* __restrict__ dv_agg, int E, int use_env) {
  __shared__ float WrS[NRBF * FDIM];
  __shared__ float brS[FDIM];
  for (int i = threadIdx.x; i < NRBF * FDIM; i += blockDim.x) WrS[i] = Wr[i];
  for (int i = threadIdx.x; i < FDIM; i += blockDim.x) brS[i] = br[i];
  __syncthreads();

  int e = blockIdx.x * blockDim.x + threadIdx.x;
  if (e >= E) return;
  int src = ei[e];
  int tgt = ei[E + e];
  float d = dArr[e];
  float rn0 = rnArr[e * 3 + 0];
  float rn1 = rnArr[e * 3 + 1];
  float rn2 = rnArr[e * 3 + 2];

  // RBF expansion: centers linspace(0, CUT, 16), width = CUT/16
  const float spacing = CUTOFF / (NRBF - 1);
  const float width   = CUTOFF / NRBF;
  const float inv2w2  = 1.0f / (2.0f * width * width);
  float rbfv[NRBF];
#pragma unroll
  for (int k = 0; k < NRBF; k++) {
    float u = d - (float)k * spacing;
    rbfv[k] = __expf(-u * u * inv2w2);
  }
  float env = 1.0f;
  if (use_env) {
    const float kPi = 3.14159265358979f;
    float dc = fminf(d, CUTOFF);
    env = 0.5f * (__cosf(kPi * dc / CUTOFF) + 1.0f);
  }

  const float4* __restrict__ phir4 = (const float4*)(phi + (size_t)src * FDIM);
  const float4* __restrict__ vrow4 = (const float4*)(vbuf + (size_t)src * 48);
  __builtin_prefetch(vrow4, 0, 3);           // global_prefetch_b8
  float* dsrow = ds_agg + (size_t)tgt * SDIM;
  float* dvrow = dv_agg + (size_t)tgt * 48;

  // ---- scalar part: ds[f] += phi[src][f] * (rbf@Wr + br)[f] * env ----
  for (int f4 = 0; f4 < SDIM / 4; f4++) {
    float4 p = phir4[f4];                    // global_load_b128
    const int f = f4 * 4;
    float fl0 = brS[f + 0], fl1 = brS[f + 1], fl2 = brS[f + 2], fl3 = brS[f + 3];
#pragma unroll
    for (int k = 0; k < NRBF; k++) {
      const float rb = rbfv[k];
      const float* w = &WrS[k * FDIM + f];
      fl0 += rb * w[0];
      fl1 += rb * w[1];
      fl2 += rb * w[2];
      fl3 += rb * w[3];
    }
    atomicAdd(&dsrow[f + 0], p.x * fl0 * env);
    atomicAdd(&dsrow[f + 1], p.y * fl1 * env);
    atomicAdd(&dsrow[f + 2], p.z * fl2 * env);
    atomicAdd(&dsrow[f + 3], p.w * fl3 * env);
  }

  // ---- vector part: preload gate values and v row as float4 ----
  float pv1[VDIM], pv2[VDIM];
#pragma unroll
  for (int q = 0; q < 4; q++) {
    float4 a = phir4[16 + q];                // phi cols 64..79
    pv1[q * 4 + 0] = a.x; pv1[q * 4 + 1] = a.y;
    pv1[q * 4 + 2] = a.z; pv1[q * 4 + 3] = a.w;
    float4 b = phir4[20 + q];                // phi cols 80..95
    pv2[q * 4 + 0] = b.x; pv2[q * 4 + 1] = b.y;
    pv2[q * 4 + 2] = b.z; pv2[q * 4 + 3] = b.w;
  }
  float vv[48];
#pragma unroll
  for (int q = 0; q < 12; q++) {
    float4 a = vrow4[q];                     // global_load_b128
    vv[q * 4 + 0] = a.x; vv[q * 4 + 1] = a.y;
    vv[q * 4 + 2] = a.z; vv[q * 4 + 3] = a.w;
  }
#pragma unroll
  for (int c = 0; c < VDIM; c++) {
    float f1 = brS[SDIM + c];
    float f2 = brS[SDIM + VDIM + c];
#pragma unroll
    for (int k = 0; k < NRBF; k++) {
      const float rb = rbfv[k];
      f1 += rb * WrS[k * FDIM + SDIM + c];
      f2 += rb * WrS[k * FDIM + SDIM + VDIM + c];
    }
    float dv1 = pv1[c] * f1 * env;
    float dv2 = pv2[c] * f2 * env;
    atomicAdd(&dvrow[0 * VDIM + c], dv1 * vv[0 * VDIM + c] + dv2 * rn0);
    atomicAdd(&dvrow[1 * VDIM + c], dv1 * vv[1 * VDIM + c] + dv2 * rn1);
    atomicAdd(&dvrow[2 * VDIM + c], dv1 * vv[2 * VDIM + c] + dv2 * rn2);
  }
}

// ---------------- fused residual + LayerNorm + v update --------------------
__global__ __launch_bounds__(256) void k_update(
    float* __restrict__ s, float* __restrict__ vbuf,
    const float* __restrict__ ds_agg, const float* __restrict__ dva_l,
    const float* __restrict__ dva_g, const float* __restrict__ cnt_l,
    const float* __restrict__ cnt_g, const float* __restrict__ ln_scale,
    const float* __restrict__ ln_bias) {
  int n = blockIdx.x * 8 + (threadIdx.x >> 5);
  if (n >= NN) return;
  int lane = threadIdx.x & 31;
  size_t base = (size_t)n * SDIM;
  float a = s[base + lane]      + ds_agg[base + lane];
  float b = s[base + 32 + lane] + ds_agg[base + 32 + lane];
  float sum = a + b;
#pragma unroll
  for (int off = 16; off >= 1; off >>= 1) sum += __shfl_xor(sum, off, 32);
  float mean = sum * (1.0f / 64.0f);
  float da = a - mean, db = b - mean;
  float vs = da * da + db * db;
#pragma unroll
  for (int off = 16; off >= 1; off >>= 1) vs += __shfl_xor(vs, off, 32);
  float rstd = rsqrtf(vs * (1.0f / 64.0f) + 1e-5f);
  s[base + lane]      = da * rstd * ln_scale[lane]      + ln_bias[lane];
  s[base + 32 + lane] = db * rstd * ln_scale[32 + lane] + ln_bias[32 + lane];

  float c1 = fmaxf(cnt_l[n], 1.0f);
  float c2 = fmaxf(cnt_g[n], 1.0f);
  float i1 = 1.0f / c1, i2 = 1.0f / c2;
  size_t vb = (size_t)n * 48;
  for (int idx = lane; idx < 48; idx += 32)
    vbuf[vb + idx] += dva_l[vb + idx] * i1 + dva_g[vb + idx] * i2;
}

// ---------------- outputs ---------------------------------------------------
__global__ void k_coords(const float* __restrict__ vbuf,
                         const float* __restrict__ Wc, float* __restrict__ out) {
  int i = blockIdx.x * blockDim.x + threadIdx.x;   // n*3 + j
  if (i >= NN * 3) return;
  const float4* vr4 = (const float4*)(vbuf + (size_t)(i / 3) * 48 + (size_t)(i % 3) * VDIM);
  float acc = 0.0f;
#pragma unroll
  for (int q = 0; q < 4; q++) {
    float4 a = vr4[q];
    acc += a.x * Wc[q * 4 + 0] + a.y * Wc[q * 4 + 1] +
           a.z * Wc[q * 4 + 2] + a.w * Wc[q * 4 + 3];
  }
  out[i] = acc;
}

// score_atoms = s @ W_atoms (64x16): one wave per 16-row tile, one col tile
__global__ __launch_bounds__(32) void k_atoms(
    const float* __restrict__ s, const float* __restrict__ Watoms,
    float* __restrict__ out) {
  const int row0 = blockIdx.x * 16;
  const int lane = threadIdx.x & 31;
  const int half = lane >> 4;
  const int l16  = lane & 15;
  v8f acc;
#pragma unroll
  for (int r = 0; r < 8; r++) acc[r] = 0.0f;
  acc = wmma_tile_f32(s + (size_t)row0 * SDIM, SDIM, Watoms, NTYPES, SDIM, acc);
#pragma unroll
  for (int r = 0; r < 8; r++)
    out[(size_t)(row0 + r + 8 * half) * NTYPES + l16] = acc[r];
}

// ---------------- host-side orchestration ----------------------------------
extern "C" void kernel_launch(void* const* d_in, const int* in_sizes, int n_in,
                              void* d_out, int out_size, void* d_ws, size_t ws_size,
                              hipStream_t stream) {
  const float* x      = (const float*)d_in[0];
  const float* t      = (const float*)d_in[1];
  const float* pos    = (const float*)d_in[2];
  const int*   eil    = (const int*)  d_in[3];
  const int*   eig    = (const int*)  d_in[4];
  const int*   batch  = (const int*)  d_in[5];
  const float* Wtime  = (const float*)d_in[6];
  const float* btime  = (const float*)d_in[7];
  const float* Watom  = (const float*)d_in[8];
  const float* batom  = (const float*)d_in[9];
  const float* Wat    = (const float*)d_in[10];
  const float* bat    = (const float*)d_in[11];
  const float* Wp_l   = (const float*)d_in[12];
  const float* bp_l   = (const float*)d_in[13];
  const float* Wr_l   = (const float*)d_in[14];
  const float* br_l   = (const float*)d_in[15];
  const float* Wp_g   = (const float*)d_in[16];
  const float* bp_g   = (const float*)d_in[17];
  const float* Wr_g   = (const float*)d_in[18];
  const float* br_g   = (const float*)d_in[19];
  const float* ln_s   = (const float*)d_in[20];
  const float* ln_b   = (const float*)d_in[21];
  const float* Wc     = (const float*)d_in[22];
  const float* Watoms = (const float*)d_in[23];

  float* W    = (float*)d_ws;
  float* sb   = W; W += (size_t)NN * SDIM;   // node scalars
  float* vb   = W; W += (size_t)NN * 48;     // node vectors (3x16)
  float* phib = W; W += (size_t)NN * FDIM;   // phi scratch
  float* dsag = W; W += (size_t)NN * SDIM;   // ds accumulator (also tmp for embed)
  float* dval = W; W += (size_t)NN * 48;     // dv accumulator (local)
  float* dvag = W; W += (size_t)NN * 48;     // dv accumulator (global)
  float* dl   = W; W += (size_t)NEL;
  float* rnl  = W; W += (size_t)NEL * 3;
  float* dg   = W; W += (size_t)NEGL;
  float* rng  = W; W += (size_t)NEGL * 3;
  float* cntl = W; W += (size_t)NN;
  float* cntg = W; W += (size_t)NN;

  // edge attrs + degrees (constant across layers)
  hipMemsetAsync(cntl, 0, 2 * (size_t)NN * sizeof(float), stream);
  k_edge_attrs<<<(NEL + 255) / 256, 256, 0, stream>>>(eil, pos, dl, rnl, cntl, NEL);
  k_edge_attrs<<<(NEGL + 255) / 256, 256, 0, stream>>>(eig, pos, dg, rng, cntg, NEGL);

  // embedding (tmp staged in dsag, which gets re-zeroed each layer)
  k_embed1<<<NN / 16, 128, 0, stream>>>(x, Watom, batom, t, batch, Wtime, btime, dsag);
  k_embed2<<<NN / 16, 128, 0, stream>>>(dsag, Wat, bat, sb);
  hipMemsetAsync(vb, 0, (size_t)NN * 48 * sizeof(float), stream);

  for (int i = 0; i < NLAYERS; i++) {
    // dsag, dval, dvag are contiguous: single memset
    hipMemsetAsync(dsag, 0, (size_t)NN * (SDIM + 96) * sizeof(float), stream);
    k_phi<<<NN / 16, 192, 0, stream>>>(sb, Wp_l + (size_t)i * SDIM * FDIM,
                                       bp_l + (size_t)i * FDIM, phib);
    k_message<<<(NEL + 255) / 256, 256, 0, stream>>>(
        eil, dl, rnl, phib, vb, Wr_l + (size_t)i * NRBF * FDIM,
        br_l + (size_t)i * FDIM, dsag, dval, NEL, 1);
    k_phi<<<NN / 16, 192, 0, stream>>>(sb, Wp_g + (size_t)i * SDIM * FDIM,
                                       bp_g + (size_t)i * FDIM, phib);
    k_message<<<(NEGL + 255) / 256, 256, 0, stream>>>(
        eig, dg, rng, phib, vb, Wr_g + (size_t)i * NRBF * FDIM,
        br_g + (size_t)i * FDIM, dsag, dvag, NEGL, 0);
    k_update<<<(NN + 7) / 8, 256, 0, stream>>>(sb, vb, dsag, dval, dvag,
                                               cntl, cntg,
                                               ln_s + (size_t)i * SDIM,
                                               ln_b + (size_t)i * SDIM);
  }

  float* out = (float*)d_out;
  k_coords<<<(NN * 3 + 255) / 256, 256, 0, stream>>>(vb, Wc, out);
  k_atoms<<<NN / 16, 32, 0, stream>>>(sb, Watoms, out + (size_t)NN * 3);
}